// ModifiedConv_22874995818645
// MI455X (gfx1250) — compile-verified
//
#include <hip/hip_runtime.h>

typedef __attribute__((ext_vector_type(2))) float v2f;
typedef __attribute__((ext_vector_type(8))) float v8f;

#define IN_F  128
#define OUT_F 64

// ---------------------------------------------------------------------------
// Kernel 1: zero the accumulation buffer (d_out)
// ---------------------------------------------------------------------------
__global__ void zero_out_kernel(float* __restrict__ out, int n) {
    int i = blockIdx.x * blockDim.x + threadIdx.x;
    if (i < n) out[i] = 0.0f;
}

// ---------------------------------------------------------------------------
// Kernel 2: h = feats @ weight   via V_WMMA_F32_16X16X4_F32 (exact fp32)
//
// Block = 128 threads = 4 waves. Each block computes a 16-row slab of h
// (16 x 64); wave w computes the 16x16 tile at columns [16w, 16w+16).
// K loop: 128 / 4 = 32 WMMA ops per tile, fully unrolled.
//
// fp32 A-matrix 16x4 layout (ISA 7.12.2): lane<16 holds M=lane, K={k,k+1} in
// {v0,v1}; lane>=16 holds M=lane-16, K={k+2,k+3}. B mirrors with M<->N.
// C/D: VGPR v holds row M = v + (lane>=16 ? 8 : 0), col N = lane&15.
// ---------------------------------------------------------------------------
__global__ __launch_bounds__(128) void gemm_wmma_f32_kernel(
    const float* __restrict__ feats,   // [N, 128] row-major
    const float* __restrict__ weight,  // [128, 64] row-major
    float* __restrict__ h,             // [N, 64]  row-major
    int n_nodes)
{
    const int lane = threadIdx.x & 31;
    const int wave = threadIdx.x >> 5;
    const bool hi  = lane >= 16;
    const int m0   = blockIdx.x * 16;
    const int n0   = wave * 16;

    // clamp (keeps EXEC all-1s for WMMA even on a ragged final block)
    int mrow = m0 + (lane & 15);
    if (mrow >= n_nodes) mrow = n_nodes - 1;
    const int ncol = n0 + (lane & 15);

    const float* __restrict__ arow = feats + (size_t)mrow * IN_F;

    v8f acc = {};
#pragma unroll
    for (int k = 0; k < IN_F; k += 4) {
        const int ka = k + (hi ? 2 : 0);
        v2f a, b;
        a.x = arow[ka];
        a.y = arow[ka + 1];
        b.x = weight[(size_t)ka * OUT_F + ncol];
        b.y = weight[(size_t)(ka + 1) * OUT_F + ncol];
        // (neg_a, A, neg_b, B, c_mod, C, reuse_a, reuse_b)
        acc = __builtin_amdgcn_wmma_f32_16x16x4_f32(
            false, a, false, b, (short)0, acc, false, false);
    }

    const int rbase = m0 + (hi ? 8 : 0);
#pragma unroll
    for (int v = 0; v < 8; ++v) {
        const int r = rbase + v;
        if (r < n_nodes) h[(size_t)r * OUT_F + ncol] = acc[v];
    }
}

// ---------------------------------------------------------------------------
// Kernel 3: for each edge e: out[dst[e], :] += h[src[e], :]
// One wave per edge: 32 lanes cover 64 features with 2 atomics each.
// Row reads are 2 x 128B fully-coalesced; h and out live in the 192MB L2,
// so gather + atomic traffic stays on-chip. src/dst scalarized so global
// ops use the SADDR (scalar base + lane offset) form.
// ---------------------------------------------------------------------------
__global__ __launch_bounds__(256) void scatter_add_kernel(
    const float* __restrict__ h,
    const int* __restrict__ src,
    const int* __restrict__ dst,
    float* __restrict__ out,
    int n_edges)
{
    const int e = blockIdx.x * (blockDim.x >> 5) + (threadIdx.x >> 5);
    if (e >= n_edges) return;
    const int lane = threadIdx.x & 31;

    const int s = __builtin_amdgcn_readfirstlane(src[e]);
    const int d = __builtin_amdgcn_readfirstlane(dst[e]);

    const float* __restrict__ hp = h + (size_t)s * OUT_F;
    float* __restrict__ op       = out + (size_t)d * OUT_F;

    const float v0 = hp[lane];
    const float v1 = hp[lane + 32];
    unsafeAtomicAdd(op + lane,      v0);   // global_atomic_add_f32
    unsafeAtomicAdd(op + lane + 32, v1);
}

// ---------------------------------------------------------------------------
// Kernel 4: out = relu(out + bias)
// ---------------------------------------------------------------------------
__global__ void finalize_kernel(float* __restrict__ out,
                                const float* __restrict__ bias, int n) {
    int i = blockIdx.x * blockDim.x + threadIdx.x;
    if (i < n) {
        float v = out[i] + bias[i & (OUT_F - 1)];
        out[i] = v > 0.0f ? v : 0.0f;
    }
}

// ---------------------------------------------------------------------------
extern "C" void kernel_launch(void* const* d_in, const int* in_sizes, int n_in,
                              void* d_out, int out_size, void* d_ws, size_t ws_size,
                              hipStream_t stream) {
    const float* feats  = (const float*)d_in[0];  // [N, 128]
    const float* weight = (const float*)d_in[1];  // [128, 64]
    const float* bias   = (const float*)d_in[2];  // [64]
    const int*   src    = (const int*)d_in[3];    // [E]
    const int*   dst    = (const int*)d_in[4];    // [E]

    float* out = (float*)d_out;                   // [N, 64]
    float* h   = (float*)d_ws;                    // [N, 64] scratch (12.8 MB)

    const int n_nodes = in_sizes[0] / IN_F;       // 50000
    const int n_edges = in_sizes[3];              // 800000
    const int n_out   = n_nodes * OUT_F;          // 3.2M floats

    // 1) zero accumulator (harness does not re-zero d_out between replays)
    zero_out_kernel<<<(n_out + 255) / 256, 256, 0, stream>>>(out, n_out);

    // 2) h = feats @ weight  (fp32 WMMA, 4 waves / block, 16x64 slab / block)
    const int m_tiles = (n_nodes + 15) / 16;      // 3125 (exact)
    gemm_wmma_f32_kernel<<<m_tiles, 128, 0, stream>>>(feats, weight, h, n_nodes);

    // 3) atomic segment-sum: one wave per edge
    const int waves_per_block = 256 / 32;
    scatter_add_kernel<<<(n_edges + waves_per_block - 1) / waves_per_block,
                         256, 0, stream>>>(h, src, dst, out, n_edges);

    // 4) relu(out + bias) in place
    finalize_kernel<<<(n_out + 255) / 256, 256, 0, stream>>>(out, bias, n_out);
}